// TransitionLayer_73770358276584
// MI455X (gfx1250) — compile-verified
//
#include <hip/hip_runtime.h>
#include <math.h>

typedef float v2f __attribute__((ext_vector_type(2)));
typedef float v8f __attribute__((ext_vector_type(8)));

#define E_CNT 256
#define M_CNT 64
#define ER_CNT 512
#define N_CNT 16
#define INV_LIT_T 1.25f  // 1/0.8
#define INV_ROW_T 2.0f   // 1/0.5

// ---------------- prep: fold selector softmax into affine (a,c) tables ----------------
// a_t/c_t stored m-major: [m][er], er = e*2 + r  (so main kernel reads float2 per e)
__global__ void __launch_bounds__(256)
prep_kernel(const float* __restrict__ sel, const float* __restrict__ alog,
            float* __restrict__ aT, float* __restrict__ cT, float* __restrict__ alpha) {
  int idx = blockIdx.x * blockDim.x + threadIdx.x;   // 0..32767
  int m  = idx >> 9;
  int er = idx & (ER_CNT - 1);
  const float* s = sel + ((size_t)er * M_CNT + m) * 3;
  float x0 = s[0] * INV_LIT_T, x1 = s[1] * INV_LIT_T, x2 = s[2] * INV_LIT_T;
  float mx = fmaxf(x0, fmaxf(x1, x2));
  float e0 = expf(x0 - mx), e1 = expf(x1 - mx), e2 = expf(x2 - mx);
  float inv = 1.f / (e0 + e1 + e2);
  float s0 = e0 * inv, s1 = e1 * inv, s2 = e2 * inv;
  aT[m * ER_CNT + er] = s0 + s2;   // L = a + c*u  ==  s0 + s1*u + s2*(1-u)
  cT[m * ER_CNT + er] = s1 - s2;
  if (idx < ER_CNT) alpha[idx] = 1.f / (1.f + expf(-alog[idx]));
}

// ---------------- main: one block per batch row b ----------------
__global__ void __launch_bounds__(256)
main_kernel(const float* __restrict__ u_t,
            const float2* __restrict__ aT2, const float2* __restrict__ cT2,
            const float2* __restrict__ alpha2, const float* __restrict__ mask,
            float* __restrict__ out) {
  const int b = blockIdx.x;
  const int e = threadIdx.x;            // edge index 0..255 (owns r=0 and r=1)

  __shared__ float lds_u[M_CNT];
  __shared__ float lds_logit[E_CNT];
  __shared__ float lds_ex[E_CNT];
  __shared__ float lds_rs[N_CNT];       // per-row reciprocal of softmax denominator

  // --- async-stage u[b, 0:64] into LDS (CDNA5 async-to-LDS path, ASYNCcnt) ---
  if (e < 16) {
    const float* gsrc = u_t + (size_t)b * M_CNT + e * 4;
    unsigned ldst = (unsigned)(unsigned long long)(&lds_u[e * 4]);
    asm volatile("global_load_async_to_lds_b128 %0, %1, off"
                 :: "v"(ldst), "v"(gsrc) : "memory");
  }
  asm volatile("s_wait_asynccnt 0" ::: "memory");
  __syncthreads();

  // --- hot loop: C[e,r] = prod_m clip(a + c*u, 0, 1), both r packed as float2 ---
  float2 acc; acc.x = 1.f; acc.y = 1.f;
  #pragma unroll 4
  for (int m = 0; m < M_CNT; ++m) {
    float um = lds_u[m];                      // LDS broadcast (uniform addr)
    float2 a = aT2[m * (ER_CNT / 2) + e];     // coalesced, L2-resident
    float2 c = cT2[m * (ER_CNT / 2) + e];
    float lx = fminf(fmaxf(fmaf(c.x, um, a.x), 0.f), 1.f);
    float ly = fminf(fmaxf(fmaf(c.y, um, a.y), 0.f), 1.f);
    acc.x *= lx; acc.y *= ly;
  }
  float2 al = alpha2[e];
  float zx = fminf(fmaxf(al.x * acc.x, 0.f), 1.f);
  float zy = fminf(fmaxf(al.y * acc.y, 0.f), 1.f);
  float g  = 1.f - (1.f - zx) * (1.f - zy);
  float logit = logf(fmaxf(g, 1e-6f)) + mask[e];
  lds_logit[e] = logit;
  __syncthreads();

  // --- masked, numerically stable row softmax (temp 0.5) ---
  const int q   = e >> 4;
  const int col = e & 15;
  bool any_fin = false;
  float mx = -__builtin_inff();
  #pragma unroll
  for (int j = 0; j < 16; ++j) {
    float v = lds_logit[q * 16 + j];
    if (fabsf(v) < __builtin_inff()) { any_fin = true; mx = fmaxf(mx, v); }
  }
  float x = any_fin ? logit : (col == q ? 10.f : 0.f);
  if (!any_fin) mx = 10.f;
  float ex = expf(INV_ROW_T * (x - mx));      // -inf logits -> exp(-inf) = 0
  lds_ex[e] = ex;
  __syncthreads();

  // --- exact f32 WMMA rowsum: D = Ex(16x16) * ones + C via 4x v_wmma_f32_16x16x4_f32 ---
  if (threadIdx.x < 32) {                      // wave 0, EXEC all ones
    const int L    = threadIdx.x;
    const int row  = L & 15;                   // A layout: lanes 0-15 / 16-31 both M=0..15
    const int koff = (L >> 4) << 1;            // lanes 16-31 carry K+2, K+3
    v8f csum = {0.f, 0.f, 0.f, 0.f, 0.f, 0.f, 0.f, 0.f};
    v2f ones; ones.x = 1.f; ones.y = 1.f;      // all-ones B is layout-invariant
    #pragma unroll
    for (int s = 0; s < 4; ++s) {
      v2f a;
      a.x = lds_ex[row * 16 + 4 * s + koff + 0];
      a.y = lds_ex[row * 16 + 4 * s + koff + 1];
      csum = __builtin_amdgcn_wmma_f32_16x16x4_f32(false, a, false, ones,
                                                   (short)0, csum, false, false);
    }
    if ((L & 15) == 0) {                       // lanes 0/16 own rows 0-7 / 8-15
      int half = L >> 4;
      #pragma unroll
      for (int r = 0; r < 8; ++r)
        lds_rs[r + 8 * half] = __builtin_amdgcn_rcpf(csum[r]);  // sum >= 1, v_rcp safe
    }
  }
  __syncthreads();

  out[(size_t)b * E_CNT + e] = ex * lds_rs[q];  // single v_mul epilogue per thread
}

// ---------------- launch ----------------
extern "C" void kernel_launch(void* const* d_in, const int* in_sizes, int n_in,
                              void* d_out, int out_size, void* d_ws, size_t ws_size,
                              hipStream_t stream) {
  const float* u_t  = (const float*)d_in[0];   // (B, 64)
  const float* sel  = (const float*)d_in[1];   // (256, 2, 64, 3)
  const float* alog = (const float*)d_in[2];   // (256, 2)
  const float* mask = (const float*)d_in[3];   // (16, 16)
  const int B = in_sizes[0] / M_CNT;

  float* ws    = (float*)d_ws;
  float* aT    = ws;               // 32768 floats
  float* cT    = ws + 32768;       // 32768 floats
  float* alpha = ws + 65536;       // 512 floats

  prep_kernel<<<(ER_CNT * M_CNT) / 256, 256, 0, stream>>>(sel, alog, aT, cT, alpha);
  main_kernel<<<B, 256, 0, stream>>>(u_t, (const float2*)aT, (const float2*)cT,
                                     (const float2*)alpha, mask, (float*)d_out);
}